// SGC_4501125726313
// MI455X (gfx1250) — compile-verified
//
#include <hip/hip_runtime.h>
#include <hip/hip_bf16.h>

typedef float v2f __attribute__((ext_vector_type(2)));
typedef float v8f __attribute__((ext_vector_type(8)));

#define SGC_BN_EPS 1e-5f

// ---------------------------------------------------------------------------
// Degree / normalization prep
// ---------------------------------------------------------------------------
__global__ void sgc_deg_init(float* __restrict__ deg, int N) {
  int i = blockIdx.x * blockDim.x + threadIdx.x;
  if (i < N) deg[i] = 1.0f;  // self-loop contributes 1
}

__global__ void sgc_deg_count(const int* __restrict__ dst, float* __restrict__ deg, int E) {
  int e = blockIdx.x * blockDim.x + threadIdx.x;
  if (e < E)
    __hip_atomic_fetch_add(&deg[dst[e]], 1.0f, __ATOMIC_RELAXED, __HIP_MEMORY_SCOPE_AGENT);
}

__global__ void sgc_deg_to_dinv(float* __restrict__ deg, int N) {
  int i = blockIdx.x * blockDim.x + threadIdx.x;
  if (i < N) deg[i] = rsqrtf(deg[i]);  // deg >= 1 always (self loops)
}

// Fold BatchNorm (eval) into the first linear: W1s[:,j] = W1[:,j]*s[j],
// b1s[j] = beta[j] + s[j]*(b1[j] - mean[j]),  s[j] = gamma[j]*rsqrt(var[j]+eps)
__global__ void sgc_wprep(const float* __restrict__ W1, const float* __restrict__ b1,
                          const float* __restrict__ gamma, const float* __restrict__ beta,
                          const float* __restrict__ bn_mean, const float* __restrict__ bn_var,
                          float* __restrict__ W1s, float* __restrict__ b1s) {
  __shared__ float s1[64];
  int t = threadIdx.x;
  if (t < 64) {
    float s = gamma[t] * rsqrtf(bn_var[t] + SGC_BN_EPS);
    s1[t] = s;
    b1s[t] = beta[t] + s * (b1[t] - bn_mean[t]);
  }
  __syncthreads();
  for (int idx = t; idx < 64 * 64; idx += 256) {
    int j = idx & 63;
    W1s[idx] = W1[idx] * s1[j];
  }
}

// ---------------------------------------------------------------------------
// Propagation: hout = D^-1/2 (A+I) D^-1/2 @ hin   (64-wide features)
// seed: self-loop term  hout[i] = hin[i] * dinv[i]^2   (also inits hout)
// ---------------------------------------------------------------------------
__global__ void sgc_seed(const float* __restrict__ hin, const float* __restrict__ dinv,
                         float* __restrict__ hout, int N) {
  int t = blockIdx.x * blockDim.x + threadIdx.x;
  if (t >= N * 16) return;
  int i = t >> 4, c4 = t & 15;
  float w = dinv[i];
  w = w * w;
  float4 v = ((const float4*)(hin + (size_t)i * 64))[c4];
  v.x *= w; v.y *= w; v.z *= w; v.w *= w;
  ((float4*)(hout + (size_t)i * 64))[c4] = v;
}

// One wave32 per edge; lane l moves columns {2l, 2l+1} as a float2.
// 256B coalesced gather from hin[src], f32 atomic scatter into hout[dst].
__global__ void sgc_prop(const float* __restrict__ hin, float* __restrict__ hout,
                         const int* __restrict__ src, const int* __restrict__ dst,
                         const float* __restrict__ dinv, int E) {
  int gt = blockIdx.x * blockDim.x + threadIdx.x;
  int e = gt >> 5;
  int lane = gt & 31;
  if (e >= E) return;
  int s = src[e];
  int d = dst[e];
  float w = dinv[s] * dinv[d];
  float2 v = ((const float2*)(hin + (size_t)s * 64))[lane];
  float* o = hout + (size_t)d * 64 + lane * 2;
  __hip_atomic_fetch_add(o,     v.x * w, __ATOMIC_RELAXED, __HIP_MEMORY_SCOPE_AGENT);
  __hip_atomic_fetch_add(o + 1, v.y * w, __ATOMIC_RELAXED, __HIP_MEMORY_SCOPE_AGENT);
}

// ---------------------------------------------------------------------------
// Fused tail: out = relu(h2 @ W1s + b1s) @ W2pad + b2, f32 WMMA 16x16x4.
// Block = 128 thr (4 waves), 64 rows/block; each wave owns a 16-row stripe.
// sAct is wave-private -> no barrier needed between the two GEMMs.
// A-fragment (16x4 f32): lane l -> row l&15, VGPR0/1 = K {k0+2*(l>>4), +1}.
// C/D (16x16 f32): VGPR v -> row v + 8*(l>>4), col l&15.
// ---------------------------------------------------------------------------
__global__ void __launch_bounds__(128) sgc_gemm_fused(
    const float* __restrict__ h2, const float* __restrict__ W1s,
    const float* __restrict__ b1s, const float* __restrict__ W2,
    const float* __restrict__ b2, float* __restrict__ out, int N) {
  __shared__ __align__(16) float sH[64][68];    // pad 68: conflict-free A loads
  __shared__ __align__(16) float sAct[64][68];
  __shared__ __align__(16) float sW1[64 * 64];
  __shared__ __align__(16) float sW2[64][48];   // W2 padded 40 -> 48 cols
  __shared__ float sB1[64];
  __shared__ float sB2[48];

  const int tid = threadIdx.x;
  const long rowBase = (long)blockIdx.x * 64;

  for (int f = tid; f < 64 * 16; f += 128) {      // h2 tile: 64x64 f32
    int r = f >> 4, c4 = f & 15;
    long gr = rowBase + r;
    if (gr >= N) gr = N - 1;                      // clamp (dup rows, stores guarded)
    float4 v = ((const float4*)(h2 + gr * 64))[c4];
    *(float4*)&sH[r][c4 * 4] = v;
  }
  for (int f = tid; f < 1024; f += 128)
    ((float4*)sW1)[f] = ((const float4*)W1s)[f];
  for (int t = tid; t < 64 * 48; t += 128) {
    int k = t / 48, j = t - k * 48;
    sW2[k][j] = (j < 40) ? W2[k * 40 + j] : 0.0f;
  }
  if (tid < 64) sB1[tid] = b1s[tid];
  if (tid < 48) sB2[tid] = (tid < 40) ? b2[tid] : 0.0f;
  __syncthreads();

  const int wv   = tid >> 5;     // wave id -> row stripe
  const int lane = tid & 31;
  const int hi   = lane >> 4;    // half-wave
  const int n    = lane & 15;    // tile col (B/C/D) and A row
  const int rs   = wv * 16;      // stripe base row in block

  // ---- GEMM1: act = relu(h2 @ W1s + b1s), 4 column tiles of 16 ----
  for (int j = 0; j < 4; ++j) {
    v8f acc;
    const float bias = sB1[16 * j + n];
    for (int v = 0; v < 8; ++v) acc[v] = bias;
    for (int k0 = 0; k0 < 64; k0 += 4) {
      const int kk = k0 + 2 * hi;
      v2f a, b;
      a[0] = sH[rs + n][kk];
      a[1] = sH[rs + n][kk + 1];
      b[0] = sW1[kk * 64 + 16 * j + n];
      b[1] = sW1[(kk + 1) * 64 + 16 * j + n];
#if defined(__HIP_DEVICE_COMPILE__)
      acc = __builtin_amdgcn_wmma_f32_16x16x4_f32(false, a, false, b, (short)0, acc,
                                                  false, false);
#endif
    }
    for (int v = 0; v < 8; ++v) {
      float val = acc[v];
      sAct[rs + v + 8 * hi][16 * j + n] = val > 0.0f ? val : 0.0f;
    }
  }

  // ---- GEMM2: out = act @ W2pad + b2, 3 column tiles (cols 40..47 dropped) ----
  for (int j = 0; j < 3; ++j) {
    v8f acc;
    const float bias = sB2[16 * j + n];
    for (int v = 0; v < 8; ++v) acc[v] = bias;
    for (int k0 = 0; k0 < 64; k0 += 4) {
      const int kk = k0 + 2 * hi;
      v2f a, b;
      a[0] = sAct[rs + n][kk];
      a[1] = sAct[rs + n][kk + 1];
      b[0] = sW2[kk][16 * j + n];
      b[1] = sW2[kk + 1][16 * j + n];
#if defined(__HIP_DEVICE_COMPILE__)
      acc = __builtin_amdgcn_wmma_f32_16x16x4_f32(false, a, false, b, (short)0, acc,
                                                  false, false);
#endif
    }
    const int col = 16 * j + n;
    if (col < 40) {
      for (int v = 0; v < 8; ++v) {
        long gr = rowBase + rs + v + 8 * hi;
        if (gr < N) out[gr * 40 + col] = acc[v];
      }
    }
  }
}

// ---------------------------------------------------------------------------
// Host-side orchestration
// ---------------------------------------------------------------------------
extern "C" void kernel_launch(void* const* d_in, const int* in_sizes, int n_in,
                              void* d_out, int out_size, void* d_ws, size_t ws_size,
                              hipStream_t stream) {
  const float* x       = (const float*)d_in[0];
  const int*   ei      = (const int*)d_in[1];
  const float* W1      = (const float*)d_in[2];
  const float* b1      = (const float*)d_in[3];
  const float* gamma   = (const float*)d_in[4];
  const float* beta    = (const float*)d_in[5];
  const float* bn_mean = (const float*)d_in[6];
  const float* bn_var  = (const float*)d_in[7];
  const float* W2      = (const float*)d_in[8];
  const float* b2      = (const float*)d_in[9];
  float* out = (float*)d_out;

  const int N = in_sizes[0] / 64;
  const int E = in_sizes[1] / 2;
  const int* src = ei;
  const int* dst = ei + E;

  float* ws   = (float*)d_ws;
  float* dinv = ws;                          // N (deg then rsqrt in place)
  float* h1   = ws + (((size_t)N + 3) & ~(size_t)3);
  float* h2   = h1 + (size_t)N * 64;
  float* W1s  = h2 + (size_t)N * 64;
  float* b1s  = W1s + 64 * 64;

  const int thr = 256;
  const int gN  = (N + thr - 1) / thr;
  const int gE  = (E + thr - 1) / thr;
  const int gNC = (N * 16 + thr - 1) / thr;            // seed: N*16 threads
  const long pth = (long)E * 32;                       // prop: one wave per edge
  const int gP  = (int)((pth + thr - 1) / thr);

  sgc_deg_init<<<gN, thr, 0, stream>>>(dinv, N);
  sgc_deg_count<<<gE, thr, 0, stream>>>(dst, dinv, E);
  sgc_deg_to_dinv<<<gN, thr, 0, stream>>>(dinv, N);
  sgc_wprep<<<1, 256, 0, stream>>>(W1, b1, gamma, beta, bn_mean, bn_var, W1s, b1s);

  // hop 1: h1 = Ahat @ x
  sgc_seed<<<gNC, thr, 0, stream>>>(x, dinv, h1, N);
  sgc_prop<<<gP, thr, 0, stream>>>(x, h1, src, dst, dinv, E);
  // hop 2: h2 = Ahat @ h1
  sgc_seed<<<gNC, thr, 0, stream>>>(h1, dinv, h2, N);
  sgc_prop<<<gP, thr, 0, stream>>>(h1, h2, src, dst, dinv, E);

  // fused linear + BN + relu + linear
  sgc_gemm_fused<<<(N + 63) / 64, 128, 0, stream>>>(h2, W1s, b1s, W2, b2, out, N);
}